// TopographicalRNN_36636071035135
// MI455X (gfx1250) — compile-verified
//
#include <hip/hip_runtime.h>

// Problem constants (match reference exactly)
#define NN   45000          // H_SHEET*W_SHEET neurons
#define KSYN 33             // K+1 entries per column
#define NNZ  (NN * KSYN)    // 1,485,000
#define TSTEPS 100
#define BATCH 32

#define KCHUNKS 50
#define KSTEP   (NN / KCHUNKS)   // 900, divisible by 4

typedef __attribute__((ext_vector_type(2))) float v2f;
typedef __attribute__((ext_vector_type(8))) float v8f;

// packed CSR entry: one uniform (scalar-cache) b64 load per synapse in the hot loop
struct __align__(8) Ent { float v; int c; };

// ---------------- transpose x (B,N) -> h (N,B) ----------------
__global__ void transpose_xk(const float* __restrict__ x, float* __restrict__ h) {
  int tid = blockIdx.x * blockDim.x + threadIdx.x;
  if (tid >= NN * BATCH) return;
  int n = tid >> 5, b = tid & 31;
  h[tid] = x[(size_t)b * NN + n];           // coalesced write
}

// ---------------- CSR build: histogram / scan / scatter ----------------
__global__ void zero_i32(int* __restrict__ p, int n) {
  int i = blockIdx.x * blockDim.x + threadIdx.x;
  if (i < n) p[i] = 0;
}

__global__ void hist_k(const int* __restrict__ rows, int* __restrict__ counts) {
  int e = blockIdx.x * blockDim.x + threadIdx.x;
  if (e < NNZ) atomicAdd(&counts[rows[e]], 1);
}

// single-workgroup exclusive scan with running carry (45000 elems, 44 chunks)
__global__ void scan_k(const int* __restrict__ in, int* __restrict__ out, int n) {
  __shared__ int buf[1024];
  __shared__ int carry_s;
  int tid = threadIdx.x;
  if (tid == 0) carry_s = 0;
  __syncthreads();
  for (int base = 0; base < n; base += 1024) {
    int i = base + tid;
    int v = (i < n) ? in[i] : 0;
    buf[tid] = v;
    __syncthreads();
    for (int off = 1; off < 1024; off <<= 1) {
      int t = (tid >= off) ? buf[tid - off] : 0;
      __syncthreads();
      buf[tid] += t;
      __syncthreads();
    }
    int incl  = buf[tid];
    int carry = carry_s;
    if (i < n) out[i] = carry + incl - v;   // exclusive
    __syncthreads();
    if (tid == 1023) carry_s = carry + incl;
    __syncthreads();
  }
  if (tid == 0) out[n] = carry_s;
}

__global__ void copy_i32(const int* __restrict__ s, int* __restrict__ d, int n) {
  int i = blockIdx.x * blockDim.x + threadIdx.x;
  if (i < n) d[i] = s[i];
}

__global__ void scatter_k(const int* __restrict__ rows, const int* __restrict__ cols,
                          const float* __restrict__ vals, int* __restrict__ offs,
                          Ent* __restrict__ ents) {
  int e = blockIdx.x * blockDim.x + threadIdx.x;
  if (e >= NNZ) return;
  int pos = atomicAdd(&offs[rows[e]], 1);
  Ent en;
  en.v = vals[e];
  en.c = cols[e];
  ents[pos] = en;                            // single b64 store
}

// ---------------- timestep: one wave32 per row, lane = batch ----------------
// row is wave-uniform; readfirstlane proves it to the compiler so row_ptr/bias/
// entry loads scalarize to s_load_b32/b64 (KM path) and the only VMEM op per
// entry is the coalesced 128B h row load in SGPR-base + lane-offset form.
__global__ __launch_bounds__(256)
void step_k(const float* __restrict__ h_in, float* __restrict__ h_out,
            const int* __restrict__ row_ptr, const Ent* __restrict__ ents,
            const float* __restrict__ bias) {
  int row  = __builtin_amdgcn_readfirstlane(
                 (int)((blockIdx.x * blockDim.x + threadIdx.x) >> 5));
  int lane = threadIdx.x & 31;
  if (row >= NN) return;
  int s = row_ptr[row];
  int e = row_ptr[row + 1];
  const Ent* __restrict__ ep = ents + s;
  int n = e - s;
  float acc = 0.f;
#pragma clang loop unroll_count(4)
  for (int i = 0; i < n; ++i) {
    Ent en = ep[i];                         // uniform -> s_load_b64
    acc = fmaf(en.v, h_in[(size_t)en.c * 32 + lane], acc);  // coalesced 128B row
  }
  float y = acc + bias[row];
  h_out[(size_t)row * 32 + lane] = y > 0.f ? y : 0.f;
}

// ---------------- head GEMM1: (32 x 45000) @ (45000 x 64), WMMA f32, K-split ----------------
__global__ __launch_bounds__(256)
void gemm1_k(const float* __restrict__ h /* (N,32): h[n*32+b] */,
             const float* __restrict__ w1 /* (N,64) */,
             float* __restrict__ partial /* (KCHUNKS, 32, 64) */) {
  int chunk = blockIdx.x;
  int kbeg  = chunk * KSTEP;
  int kend  = kbeg + KSTEP;

  int w    = threadIdx.x >> 5;              // wave 0..7
  int lane = threadIdx.x & 31;
  int mt   = (w & 1) * 16;                  // M tile base (batch)
  int nt   = (w >> 1) * 16;                 // N tile base (hidden)
  int half = lane >> 4;
  int lm   = lane & 15;
  int m = mt + lm;                          // A row for this lane
  int j = nt + lm;                          // B col for this lane
  v8f c = {};
  for (int k = kbeg; k < kend; k += 4) {
    int k0 = k + (half ? 2 : 0);
    int k1 = k + (half ? 3 : 1);
    v2f a, b;
    a.x = h[(size_t)k0 * 32 + m];           // A[m][k0]  (A = h^T)
    a.y = h[(size_t)k1 * 32 + m];           // A[m][k1]
    b.x = w1[(size_t)k0 * 64 + j];          // B[k0][j]
    b.y = w1[(size_t)k1 * 64 + j];          // B[k1][j]
    c = __builtin_amdgcn_wmma_f32_16x16x4_f32(
        /*neg_a=*/false, a, /*neg_b=*/false, b,
        /*c_mod=*/(short)0, c, /*reuse_a=*/false, /*reuse_b=*/false);
  }
  // D layout: VGPR v holds rows (v) and (v+8) of the 16x16 tile
  int mrow = mt + (half ? 8 : 0);
  float* pt = partial + (size_t)chunk * 32 * 64;
#pragma unroll
  for (int v = 0; v < 8; ++v)
    pt[(size_t)(mrow + v) * 64 + j] = c[v];
}

// deterministic reduction over K-chunks + bias + ReLU
__global__ void reduce1_k(const float* __restrict__ partial,
                          const float* __restrict__ b1, float* __restrict__ a1) {
  int o = blockIdx.x * blockDim.x + threadIdx.x;  // 0..2047
  if (o >= 32 * 64) return;
  float s = 0.f;
#pragma unroll 5
  for (int c = 0; c < KCHUNKS; ++c) s += partial[(size_t)c * 32 * 64 + o];
  s += b1[o & 63];
  a1[o] = s > 0.f ? s : 0.f;
}

// ---------------- head GEMM2: (32 x 64) @ (64 x 10) ----------------
__global__ void gemm2_k(const float* __restrict__ a1, const float* __restrict__ w2,
                        const float* __restrict__ b2, float* __restrict__ out) {
  int tid = blockIdx.x * blockDim.x + threadIdx.x;
  if (tid >= 32 * 10) return;
  int b = tid / 10, o = tid % 10;
  float s = b2[o];
#pragma unroll
  for (int j = 0; j < 64; ++j)
    s = fmaf(a1[b * 64 + j], w2[j * 10 + o], s);
  out[tid] = s;
}

// ---------------- driver ----------------
extern "C" void kernel_launch(void* const* d_in, const int* in_sizes, int n_in,
                              void* d_out, int out_size, void* d_ws, size_t ws_size,
                              hipStream_t stream) {
  const float* x    = (const float*)d_in[0];   // (B,N)
  const float* vals = (const float*)d_in[1];   // (nnz)
  const float* bias = (const float*)d_in[2];   // (N)
  const float* w1   = (const float*)d_in[3];   // (N,64)
  const float* b1   = (const float*)d_in[4];   // (64)
  const float* w2   = (const float*)d_in[5];   // (64,10)
  const float* b2   = (const float*)d_in[6];   // (10)
  const int*   rows = (const int*)d_in[7];     // (nnz)
  const int*   cols = (const int*)d_in[8];     // (nnz)
  float* out = (float*)d_out;                  // (32,10)

  // workspace carve-out (~24.5 MB), 256B aligned slabs
  char* ws = (char*)d_ws;
  size_t off = 0;
  auto carve = [&](size_t bytes) -> void* {
    void* p = ws + off;
    off = (off + bytes + 255) & ~(size_t)255;
    return p;
  };
  float* hA      = (float*)carve((size_t)NN * BATCH * 4);
  float* hB      = (float*)carve((size_t)NN * BATCH * 4);
  int*   row_ptr = (int*)  carve((size_t)(NN + 1) * 4);
  int*   offs    = (int*)  carve((size_t)NN * 4);
  Ent*   ents    = (Ent*)  carve((size_t)NNZ * 8);
  float* partial = (float*)carve((size_t)KCHUNKS * 32 * 64 * 4);
  float* a1      = (float*)carve((size_t)32 * 64 * 4);

  // CSR-by-row build (amortized over 100 timesteps)
  zero_i32 <<<(NN  + 255) / 256, 256, 0, stream>>>(offs, NN);
  hist_k   <<<(NNZ + 255) / 256, 256, 0, stream>>>(rows, offs);
  scan_k   <<<1, 1024, 0, stream>>>(offs, row_ptr, NN);
  copy_i32 <<<(NN  + 255) / 256, 256, 0, stream>>>(row_ptr, offs, NN);
  scatter_k<<<(NNZ + 255) / 256, 256, 0, stream>>>(rows, cols, vals, offs, ents);

  // h0 = x^T  (N,B)
  transpose_xk<<<((size_t)NN * BATCH + 255) / 256, 256, 0, stream>>>(x, hA);

  // 100 dependent timesteps, ping-pong buffers; all L2-resident (~24 MB << 192 MB)
  const float* hin = hA;
  float* hout = hB;
  const int step_blocks = (NN * 32 + 255) / 256;   // one wave32 per row
  for (int t = 0; t < TSTEPS; ++t) {
    step_k<<<step_blocks, 256, 0, stream>>>(hin, hout, row_ptr, ents, bias);
    const float* tmp = hin; hin = hout; hout = (float*)tmp;
  }

  // dense head: K-split WMMA + deterministic reduce
  gemm1_k <<<KCHUNKS, 256, 0, stream>>>(hin, w1, partial);
  reduce1_k<<<(32 * 64 + 255) / 256, 256, 0, stream>>>(partial, b1, a1);
  gemm2_k <<<1, 320, 0, stream>>>(a1, w2, b2, out);
}